// CrystalGNN_81406810129006
// MI455X (gfx1250) — compile-verified
//
#include <hip/hip_runtime.h>
#include <math.h>

// ---------------------------------------------------------------------------
// CrystalGNN on MI455X (gfx1250): GCNConv (fp32 WMMA) + scatter + pool + head
// ---------------------------------------------------------------------------

#define NUM_NODES   10000
#define NUM_FEAT    128
#define HIDDEN      128
#define NUM_GRAPHS  64

typedef __attribute__((ext_vector_type(2))) float v2f;
typedef __attribute__((ext_vector_type(8))) float v8f;

// ---- 1. init: deg = 1 (self loop), pool = 0, cnt = 0 ----------------------
__global__ __launch_bounds__(256) void init_kernel(float* __restrict__ deg,
                                                   float* __restrict__ pool,
                                                   float* __restrict__ cnt,
                                                   int n) {
    int i = blockIdx.x * blockDim.x + threadIdx.x;
    if (i < n)                      deg[i]  = 1.0f;
    if (i < NUM_GRAPHS * HIDDEN)    pool[i] = 0.0f;
    if (i < NUM_GRAPHS)             cnt[i]  = 0.0f;
}

// ---- 2. in-degree accumulation over edges ---------------------------------
__global__ __launch_bounds__(256) void deg_kernel(const int* __restrict__ dst,
                                                  float* __restrict__ deg, int E) {
    int e = blockIdx.x * blockDim.x + threadIdx.x;
    if (e < E) atomicAdd(&deg[dst[e]], 1.0f);
}

// ---- 3. deg -> 1/sqrt(deg) in place (deg >= 1 always, self loops) ---------
__global__ __launch_bounds__(256) void rsqrt_kernel(float* __restrict__ deg, int n) {
    int i = blockIdx.x * blockDim.x + threadIdx.x;
    if (i < n) deg[i] = rsqrtf(deg[i]);
}

// ---- 4. h = x @ W via V_WMMA_F32_16X16X4_F32 ------------------------------
// One wave per 16x16 output tile; K=128 in steps of 4.
// A-frag (16x4 f32): lane<16 -> row=lane, K={0,1}; lane>=16 -> K={2,3}.
// B-frag (4x16 f32): mirrored K split across lane halves.
// C/D  (16x16 f32): VGPR r -> rows r (lanes 0-15) / r+8 (lanes 16-31).
__global__ __launch_bounds__(256) void gemm_kernel(const float* __restrict__ x,
                                                   const float* __restrict__ W,
                                                   float* __restrict__ h,
                                                   int tilesM) {
    const int wave = threadIdx.x >> 5;            // 8 waves / block
    const int lane = threadIdx.x & 31;
    const int tile = blockIdx.x * 8 + wave;
    const int tm   = tile >> 3;                   // HIDDEN/16 == 8 tiles in N
    const int tn   = tile & 7;
    if (tm >= tilesM) return;                     // wave-uniform (never splits EXEC)

    const int lid  = lane & 15;
    const int koff = (lane >> 4) * 2;             // 0 or 2
    const int row  = tm * 16 + lid;
    const int col  = tn * 16 + lid;
    const float* __restrict__ xrow = x + (long)row * NUM_FEAT;

    v8f acc = {};
    for (int k = 0; k < NUM_FEAT; k += 4) {
        v2f a = *(const v2f*)(xrow + k + koff);   // 8B-aligned
        v2f b;
        b.x = W[(k + koff)     * HIDDEN + col];
        b.y = W[(k + koff + 1) * HIDDEN + col];
        acc = __builtin_amdgcn_wmma_f32_16x16x4_f32(
                  false, a, false, b, (short)0, acc, false, false);
    }

    const int srow = tm * 16 + (lane >> 4) * 8;
    float* __restrict__ hp = h + (long)srow * HIDDEN + col;
    #pragma unroll
    for (int r = 0; r < 8; ++r)
        hp[r * HIDDEN] = acc[r];
}

// ---- 5. out = dinv^2 * h (self loop) + bias -------------------------------
__global__ __launch_bounds__(256) void selfloop_kernel(const float* __restrict__ h,
                                                       const float* __restrict__ dinv,
                                                       const float* __restrict__ b,
                                                       float* __restrict__ out, int n) {
    int i = blockIdx.x * blockDim.x + threadIdx.x;
    if (i >= n * HIDDEN) return;
    int node = i >> 7;                // /HIDDEN
    int f    = i & (HIDDEN - 1);
    float di = dinv[node];
    out[i] = di * di * h[i] + b[f];
}

// ---- 6. edge scatter: out[dst] += dinv[src]*dinv[dst] * h[src] ------------
// One wave per edge; each lane handles 4 consecutive features (float4 load,
// 4x global_atomic_add_f32 into L2-resident out).
__global__ __launch_bounds__(256) void scatter_kernel(const int* __restrict__ src,
                                                      const int* __restrict__ dst,
                                                      const float* __restrict__ dinv,
                                                      const float* __restrict__ h,
                                                      float* __restrict__ out, int E) {
    int wave = (blockIdx.x * blockDim.x + threadIdx.x) >> 5;
    int lane = threadIdx.x & 31;
    if (wave >= E) return;
    int s = src[wave];
    int d = dst[wave];
    float norm = dinv[s] * dinv[d];
    const float4 hv = *(const float4*)(h + (long)s * HIDDEN + lane * 4);
    float* op = out + (long)d * HIDDEN + lane * 4;
    atomicAdd(op + 0, norm * hv.x);
    atomicAdd(op + 1, norm * hv.y);
    atomicAdd(op + 2, norm * hv.z);
    atomicAdd(op + 3, norm * hv.w);
}

// ---- 7. ReLU + mean-pool accumulation -------------------------------------
__global__ __launch_bounds__(256) void pool_kernel(const float* __restrict__ agg,
                                                   const int* __restrict__ batch,
                                                   float* __restrict__ pool,
                                                   float* __restrict__ cnt, int n) {
    int wave = (blockIdx.x * blockDim.x + threadIdx.x) >> 5;
    int lane = threadIdx.x & 31;
    if (wave >= n) return;
    int g = batch[wave];
    const float4 v = *(const float4*)(agg + (long)wave * HIDDEN + lane * 4);
    float* pp = pool + (long)g * HIDDEN + lane * 4;
    atomicAdd(pp + 0, fmaxf(v.x, 0.0f));
    atomicAdd(pp + 1, fmaxf(v.y, 0.0f));
    atomicAdd(pp + 2, fmaxf(v.z, 0.0f));
    atomicAdd(pp + 3, fmaxf(v.w, 0.0f));
    if (lane == 0) atomicAdd(&cnt[g], 1.0f);
}

// ---- 8. head: mean, FC(128->2), log_softmax -------------------------------
__global__ __launch_bounds__(64) void head_kernel(const float* __restrict__ pool,
                                                  const float* __restrict__ cnt,
                                                  const float* __restrict__ Wfc,
                                                  const float* __restrict__ bfc,
                                                  float* __restrict__ out) {
    int g = threadIdx.x;
    if (g >= NUM_GRAPHS) return;
    float inv = 1.0f / fmaxf(cnt[g], 1.0f);
    float l0 = bfc[0], l1 = bfc[1];
    const float* pg = pool + (long)g * HIDDEN;
    for (int f = 0; f < HIDDEN; ++f) {
        float p = pg[f] * inv;
        l0 += p * Wfc[f * 2 + 0];
        l1 += p * Wfc[f * 2 + 1];
    }
    float m   = fmaxf(l0, l1);
    float lse = m + logf(expf(l0 - m) + expf(l1 - m));
    out[g * 2 + 0] = l0 - lse;
    out[g * 2 + 1] = l1 - lse;
}

// ---------------------------------------------------------------------------
extern "C" void kernel_launch(void* const* d_in, const int* in_sizes, int n_in,
                              void* d_out, int out_size, void* d_ws, size_t ws_size,
                              hipStream_t stream) {
    const float* x     = (const float*)d_in[0];   // [N, 128]
    const int*   eidx  = (const int*)  d_in[1];   // [2, E]
    const int*   batch = (const int*)  d_in[2];   // [N]
    const float* W     = (const float*)d_in[3];   // [128, 128]
    const float* b     = (const float*)d_in[4];   // [128]
    const float* Wfc   = (const float*)d_in[5];   // [128, 2]
    const float* bfc   = (const float*)d_in[6];   // [2]
    float*       out   = (float*)d_out;           // [64, 2]

    const int N = in_sizes[0] / NUM_FEAT;         // 10000
    const int E = in_sizes[1] / 2;                // 640000
    const int* src = eidx;
    const int* dst = eidx + E;

    // workspace layout (floats)
    float* ws   = (float*)d_ws;
    float* deg  = ws;                             // N (padded 10240) -> becomes dinv
    float* h    = ws + 10240;                     // N*128
    float* agg  = h  + (size_t)N * HIDDEN;        // N*128
    float* pool = agg + (size_t)N * HIDDEN;       // 64*128
    float* cnt  = pool + NUM_GRAPHS * HIDDEN;     // 64

    const int tilesM = N / 16;                    // 625
    const int nTiles = tilesM * (HIDDEN / 16);    // 5000 waves

    init_kernel    <<<(N + 255) / 256, 256, 0, stream>>>(deg, pool, cnt, N);
    deg_kernel     <<<(E + 255) / 256, 256, 0, stream>>>(dst, deg, E);
    rsqrt_kernel   <<<(N + 255) / 256, 256, 0, stream>>>(deg, N);
    gemm_kernel    <<<(nTiles + 7) / 8, 256, 0, stream>>>(x, W, h, tilesM);
    selfloop_kernel<<<(N * HIDDEN + 255) / 256, 256, 0, stream>>>(h, deg, b, agg, N);
    scatter_kernel <<<(E * 32 + 255) / 256, 256, 0, stream>>>(src, dst, deg, h, agg, E);
    pool_kernel    <<<(N * 32 + 255) / 256, 256, 0, stream>>>(agg, batch, pool, cnt, N);
    head_kernel    <<<1, 64, 0, stream>>>(pool, cnt, Wfc, bfc, out);
}